// AdaWinBlock1d_48704929136705
// MI455X (gfx1250) — compile-verified
//
#include <hip/hip_runtime.h>
#include <hip/hip_bf16.h>

typedef float v2f __attribute__((ext_vector_type(2)));
typedef float v8f __attribute__((ext_vector_type(8)));

#define LRELU_SLOPE 0.2f
#define INV_SQRT2   0.70710678118654752440f

constexpr int Bn = 16;     // batch
constexpr int Cc = 512;    // channels
constexpr int C2 = 1024;   // 2*C
constexpr int Ss = 128;    // style dim
constexpr int Tt = 2048;   // time
constexpr int HALF_W = 18; // (37-1)/2
constexpr int WLEN = 37;

// ---------------------------------------------------------------------------
// Kernel 0: mask + 1/(win_sum(mask)+eps).  mask is a prefix (t < L), so the
// 37-tap window sum is a closed-form clamp -- no memory reduction needed.
// ---------------------------------------------------------------------------
__global__ void mask_invden_kernel(const int* __restrict__ lengths,
                                   float* __restrict__ mask,
                                   float* __restrict__ invden) {
  const int b = blockIdx.x;
  const int L = lengths[b];
  for (int t = threadIdx.x; t < Tt; t += blockDim.x) {
    float m = (t < L) ? 1.0f : 0.0f;
    int lo = t - HALF_W; if (lo < 0) lo = 0;
    int hi = t + HALF_W; if (hi > Tt - 1) hi = Tt - 1;
    if (hi > L - 1) hi = L - 1;
    int cnt = hi - lo + 1; if (cnt < 0) cnt = 0;
    mask[b * Tt + t]   = m;
    invden[b * Tt + t] = 1.0f / ((float)cnt + 1e-9f);
  }
}

// ---------------------------------------------------------------------------
// Weight transpose: conv W [M][K][3] -> Wt [3][M][K] so the per-lane A-pair
// (k, k+1) becomes one aligned b64 load.
// ---------------------------------------------------------------------------
__global__ void wtrans_kernel(const float* __restrict__ W, float* __restrict__ Wt) {
  const long long n = (long long)Cc * Cc * 3;
  for (long long i = (long long)blockIdx.x * blockDim.x + threadIdx.x; i < n;
       i += (long long)gridDim.x * blockDim.x) {
    const int tap = (int)(i % 3);
    const long long mk = i / 3;                  // m*K + k
    Wt[(long long)tap * Cc * Cc + mk] = W[i];
  }
}

// ---------------------------------------------------------------------------
// WMMA fp32 GEMM with optional 3-tap shift (1D conv) and fused residual.
//   Out[b, m, t] = bias[m] + sum_{tap} sum_k W[tap][m][k] * In[b, k, t+tap-TAPS/2]
// One wave per 16(M) x 128(N) tile; K stepped by 4 via V_WMMA_F32_16X16X4_F32.
// A (16x4 f32): lanes 0-15 row m0+l hold K={0,1}, lanes 16-31 hold K={2,3},
// so each lane's A pair is one b64 load.  B (4x16): mirrored, 2 coalesced b32
// loads per lane.  C/D: vgpr r = row m0+r+8*half, col = lane&15.
// ---------------------------------------------------------------------------
template <int TAPS, bool RESID>
__global__ void wmma_gemm_kernel(const float* __restrict__ W,   // [TAPS][M][K]
                                 const float* __restrict__ bias,
                                 const float* __restrict__ In,
                                 const float* __restrict__ Xres,
                                 float* __restrict__ Out,
                                 int K, long long tapStride,
                                 long long inBatch, long long outBatch) {
  const int lane = threadIdx.x & 31;
  const int half = lane >> 4;   // selects K pair {0,1} vs {2,3}
  const int l    = lane & 15;
  const int t0 = blockIdx.x * 128;
  const int m0 = blockIdx.y * 16;
  const int b  = blockIdx.z;
  const float* Inb = In + (long long)b * inBatch;

  const v8f zero = {0.f, 0.f, 0.f, 0.f, 0.f, 0.f, 0.f, 0.f};
  v8f acc[8] = {zero, zero, zero, zero, zero, zero, zero, zero};

  for (int k0 = 0; k0 < K; k0 += 4) {
    const int ka = k0 + 2 * half;             // even -> 8B-aligned A pair
#pragma unroll
    for (int tap = 0; tap < TAPS; ++tap) {
      const float* wp = W + (long long)tap * tapStride
                          + (long long)(m0 + l) * K + ka;
      const v2f a = *(const v2f*)wp;          // one global_load_b64
      const float* ipb = Inb + (long long)ka * Tt;
#pragma unroll
      for (int nt = 0; nt < 8; ++nt) {
        const int t = t0 + nt * 16 + l + tap - (TAPS / 2);
        v2f bf;
        if (TAPS == 1) {                      // fc path: always in range
          bf.x = ipb[t];
          bf.y = ipb[t + Tt];
        } else {                              // conv path: zero-pad halo
          const bool ok = (t >= 0) && (t < Tt);
          bf.x = ok ? ipb[t]      : 0.0f;
          bf.y = ok ? ipb[t + Tt] : 0.0f;
        }
        acc[nt] = __builtin_amdgcn_wmma_f32_16x16x4_f32(
            false, a, false, bf, (short)0, acc[nt], false, false);
      }
    }
  }

  // epilogue: bias (+ residual)/sqrt2, write D per the 16x16 f32 C/D layout
#pragma unroll
  for (int nt = 0; nt < 8; ++nt) {
    const int t = t0 + nt * 16 + l;
#pragma unroll
    for (int r = 0; r < 8; ++r) {
      const int m = m0 + r + 8 * half;
      float v = acc[nt][r] + bias[m];
      if (RESID) {
        v = (v + Xres[(long long)b * outBatch + (long long)m * Tt + t]) * INV_SQRT2;
      }
      Out[(long long)b * outBatch + (long long)m * Tt + t] = v;
    }
  }
}

// ---------------------------------------------------------------------------
// AdaWin + leaky ReLU, IN PLACE into the gamma half of h.
// Block = one (b, c) row.  gamma/beta rows -> LDS with zero halo, sliding
// 37-tap running sums (8 t per thread), tanh(alpha*x) modulation.
// ---------------------------------------------------------------------------
__global__ void adawin_kernel(float* __restrict__ h,        // [B, 2C, T]
                              const float* __restrict__ xin,// [B, C, T]
                              const float* __restrict__ mask,
                              const float* __restrict__ invden,
                              const float* __restrict__ alpha_p) {
  const int c = blockIdx.x;
  const int b = blockIdx.y;
  const float alpha = alpha_p[0];

  float* hg = h + (long long)b * C2 * Tt + (long long)c * Tt;
  const float* hb = h + (long long)b * C2 * Tt + (long long)(Cc + c) * Tt;
  const float* xr = xin + (long long)b * Cc * Tt + (long long)c * Tt;
  const float* mr = mask + b * Tt;
  const float* dr = invden + b * Tt;

  __shared__ float sg[Tt + 2 * HALF_W];
  __shared__ float sb[Tt + 2 * HALF_W];
  for (int i = threadIdx.x; i < Tt + 2 * HALF_W; i += blockDim.x) {
    const int t = i - HALF_W;
    const bool in = (t >= 0) && (t < Tt);
    sg[i] = in ? hg[t] : 0.0f;
    sb[i] = in ? hb[t] : 0.0f;
  }
  __syncthreads();

  const int t_base = threadIdx.x * 8;   // 256 threads * 8 = 2048
  float sumg = 0.0f, sumb = 0.0f;
#pragma unroll
  for (int u = 0; u < WLEN; ++u) {      // window for t covers sg[t .. t+36]
    sumg += sg[t_base + u];
    sumb += sb[t_base + u];
  }

  float outv[8];
#pragma unroll
  for (int j = 0; j < 8; ++j) {
    const int t = t_base + j;
    const float scale = dr[t] * mr[t];
    const float gamma = sumg * scale;
    const float beta  = sumb * scale;
    const float xn = tanhf(alpha * xr[t]);
    float v = (1.0f + gamma) * xn + beta;
    v = (v > 0.0f) ? v : LRELU_SLOPE * v;   // fused leaky ReLU
    outv[j] = v;
    if (j < 7) {
      sumg += sg[t + WLEN] - sg[t];
      sumb += sb[t + WLEN] - sb[t];
    }
  }
  // global hg is only read during the LDS fill (before the barrier), and only
  // this block touches row (b,c): safe to overwrite in place.
#pragma unroll
  for (int j = 0; j < 8; ++j) hg[t_base + j] = outv[j];
}

// ---------------------------------------------------------------------------
extern "C" void kernel_launch(void* const* d_in, const int* in_sizes, int n_in,
                              void* d_out, int out_size, void* d_ws, size_t ws_size,
                              hipStream_t stream) {
  const float* x       = (const float*)d_in[0];
  const float* s       = (const float*)d_in[1];
  const int*   lengths = (const int*)d_in[2];
  const float* fc1_w   = (const float*)d_in[3];
  const float* fc1_b   = (const float*)d_in[4];
  const float* alpha1  = (const float*)d_in[5];
  const float* conv1_w = (const float*)d_in[6];
  const float* conv1_b = (const float*)d_in[7];
  const float* fc2_w   = (const float*)d_in[8];
  const float* fc2_b   = (const float*)d_in[9];
  const float* alpha2  = (const float*)d_in[10];
  const float* conv2_w = (const float*)d_in[11];
  const float* conv2_b = (const float*)d_in[12];
  float* out = (float*)d_out;

  // workspace (floats):
  //   h [B,2C,T] | z [B,C,T] | w1t [3,C,C] | w2t [3,C,C] | mask [B,T] | invden [B,T]
  float* ws  = (float*)d_ws;
  float* h   = ws;
  float* z   = h   + (long long)Bn * C2 * Tt;
  float* w1t = z   + (long long)Bn * Cc * Tt;
  float* w2t = w1t + (long long)3 * Cc * Cc;
  float* mk  = w2t + (long long)3 * Cc * Cc;
  float* idn = mk  + (long long)Bn * Tt;

  const long long sBatch   = (long long)Ss * Tt;
  const long long hBatch   = (long long)C2 * Tt;
  const long long cBatch   = (long long)Cc * Tt;
  const long long convTapS = (long long)Cc * Cc;

  // 0) transpose conv weights to [tap][M][K]
  wtrans_kernel<<<1536, 256, 0, stream>>>(conv1_w, w1t);
  wtrans_kernel<<<1536, 256, 0, stream>>>(conv2_w, w2t);

  // 1) mask + inverse window denominator
  mask_invden_kernel<<<Bn, 256, 0, stream>>>(lengths, mk, idn);

  // 2) h1 = fc1_w @ s + fc1_b
  wmma_gemm_kernel<1, false><<<dim3(Tt / 128, C2 / 16, Bn), 32, 0, stream>>>(
      fc1_w, fc1_b, s, nullptr, h, Ss, 0, sBatch, hBatch);

  // 3) y1 = leaky(adawin(x, h1))  (in place into gamma half of h)
  adawin_kernel<<<dim3(Cc, Bn), 256, 0, stream>>>(h, x, mk, idn, alpha1);

  // 4) z = conv1(y1)   (y1 rows live at batch-stride 2C*T inside h)
  wmma_gemm_kernel<3, false><<<dim3(Tt / 128, Cc / 16, Bn), 32, 0, stream>>>(
      w1t, conv1_b, h, nullptr, z, Cc, convTapS, hBatch, cBatch);

  // 5) h2 = fc2_w @ s + fc2_b  (h buffer free again)
  wmma_gemm_kernel<1, false><<<dim3(Tt / 128, C2 / 16, Bn), 32, 0, stream>>>(
      fc2_w, fc2_b, s, nullptr, h, Ss, 0, sBatch, hBatch);

  // 6) y2 = leaky(adawin(z, h2))  (in place)
  adawin_kernel<<<dim3(Cc, Bn), 256, 0, stream>>>(h, z, mk, idn, alpha2);

  // 7) out = (conv2(y2) + x) / sqrt(2)
  wmma_gemm_kernel<3, true><<<dim3(Tt / 128, Cc / 16, Bn), 32, 0, stream>>>(
      w2t, conv2_b, h, x, out, Cc, convTapS, hBatch, cBatch);
}